// FuzzyDirGCN_75462575391410
// MI455X (gfx1250) — compile-verified
//
#include <hip/hip_runtime.h>
#include <hip/hip_bf16.h>
#include <stdint.h>

#define D_FEAT 128
#define H_FEAT 128
#define O_FEAT 64
#define EPSV   1e-12f
#define APAD   132   // LDS row stride (floats): 132%64==4 banks -> conflict-free b64 frags

typedef float v2f __attribute__((ext_vector_type(2)));
typedef float v8f __attribute__((ext_vector_type(8)));
typedef int   v4i __attribute__((ext_vector_type(4)));

#if defined(__HIP_DEVICE_COMPILE__) && __has_builtin(__builtin_amdgcn_global_load_async_to_lds_b128)
#define HAVE_ASYNC_LDS 1
#else
#define HAVE_ASYNC_LDS 0
#endif

__device__ __forceinline__ void atomAddF(float* p, float v) {
  __hip_atomic_fetch_add(p, v, __ATOMIC_RELAXED, __HIP_MEMORY_SCOPE_AGENT);
}

__device__ __forceinline__ void async_copy16(const float* gsrc, float* ldst) {
#if HAVE_ASYNC_LDS
  typedef __attribute__((address_space(1))) v4i* gv4i_p;
  typedef __attribute__((address_space(3))) v4i* lv4i_p;
  __builtin_amdgcn_global_load_async_to_lds_b128((gv4i_p)gsrc, (lv4i_p)ldst, 0, 0);
#else
  *(float4*)ldst = *(const float4*)gsrc;
#endif
}

__device__ __forceinline__ void async_wait_all() {
#if HAVE_ASYNC_LDS
#if __has_builtin(__builtin_amdgcn_s_wait_asynccnt)
  __builtin_amdgcn_s_wait_asynccnt(0);
#else
  asm volatile("s_wait_asynccnt 0x0" ::: "memory");
#endif
#endif
}

// ---------------------------------------------------------------------------
// Phase 1a: weighted degrees (segment sums of cos^2 / sin^2 over src and dst).
// ---------------------------------------------------------------------------
__global__ void deg_kernel(const long long* __restrict__ src,
                           const long long* __restrict__ dst,
                           const float* __restrict__ theta,
                           float* __restrict__ degof, float* __restrict__ degif,
                           float* __restrict__ degob, float* __restrict__ degib,
                           int E) {
  int e = blockIdx.x * blockDim.x + threadIdx.x;
  if (e >= E) return;
  float c  = cosf(theta[e]);
  float wf = c * c;
  float wb = 1.0f - wf;
  int s = (int)src[e], d = (int)dst[e];
  atomAddF(degof + s, wf);
  atomAddF(degif + d, wf);
  atomAddF(degob + s, wb);
  atomAddF(degib + d, wb);
}

// ---------------------------------------------------------------------------
// Phase 1b: per-direction symmetric normalization of edge weights.
// ---------------------------------------------------------------------------
__global__ void wnorm_kernel(const long long* __restrict__ src,
                             const long long* __restrict__ dst,
                             const float* __restrict__ theta,
                             const float* __restrict__ degof, const float* __restrict__ degif,
                             const float* __restrict__ degob, const float* __restrict__ degib,
                             float* __restrict__ wfn, float* __restrict__ wbn, int E) {
  int e = blockIdx.x * blockDim.x + threadIdx.x;
  if (e >= E) return;
  float c  = cosf(theta[e]);
  float wf = c * c;
  float wb = 1.0f - wf;
  int s = (int)src[e], d = (int)dst[e];
  wfn[e] = wf * rsqrtf(degof[s] * degif[d] + EPSV);
  wbn[e] = wb * rsqrtf(degob[s] * degib[d] + EPSV);
}

// ---------------------------------------------------------------------------
// Phase 2: edge aggregation.  One wave per edge, one float4 per lane:
// coalesced 512B gather of x[src], weighted scatter-add (global f32 atomics,
// no-return -> STOREcnt path) into a_s2d[dst] / a_d2s[dst].  L2-resident.
// ---------------------------------------------------------------------------
__global__ void agg_kernel(const float* __restrict__ x,
                           const long long* __restrict__ src,
                           const long long* __restrict__ dst,
                           const float* __restrict__ wfn,
                           const float* __restrict__ wbn,
                           float* __restrict__ aS, float* __restrict__ aD,
                           int E) {
  long long gid = (long long)blockIdx.x * blockDim.x + threadIdx.x;
  int e = (int)(gid >> 5);
  if (e >= E) return;
  int c = ((int)gid & 31) * 4;
  int s = (int)src[e], d = (int)dst[e];
  const float4 v = *(const float4*)(x + (size_t)s * D_FEAT + c);
  float wf = wfn[e], wb = wbn[e];
  float* ps = aS + (size_t)d * D_FEAT + c;
  float* pd = aD + (size_t)d * D_FEAT + c;
  atomAddF(ps + 0, v.x * wf);
  atomAddF(ps + 1, v.y * wf);
  atomAddF(ps + 2, v.z * wf);
  atomAddF(ps + 3, v.w * wf);
  atomAddF(pd + 0, v.x * wb);
  atomAddF(pd + 1, v.y * wb);
  atomAddF(pd + 2, v.z * wb);
  atomAddF(pd + 3, v.w * wb);
}

// ---------------------------------------------------------------------------
// Phase 3: fused triple GEMM, V_WMMA_F32_16X16X4_F32 (exact fp32).
//   h = aS @ Wa + aD @ Wb + x @ Wc + (ba+bb+bc), optional ReLU.
// Block = 256 thr (8 waves) = one 16-row M tile; wave w owns N-tile w.
// The three 16x128 A tiles are staged in LDS once per block (async-to-LDS
// when available), then every wave reads A fragments via conflict-free
// ds_load_b64 (row stride 132 floats).  K=128 -> 32 ksteps x 3 = 96 WMMAs/wave.
// ---------------------------------------------------------------------------
__global__ __launch_bounds__(256) void gemm_hidden_kernel(
    const float* __restrict__ aS, const float* __restrict__ aD,
    const float* __restrict__ xin,
    const float* __restrict__ Wa, const float* __restrict__ Wb,
    const float* __restrict__ Wc,
    const float* __restrict__ ba, const float* __restrict__ bb,
    const float* __restrict__ bc,
    float* __restrict__ out, int N, int do_relu) {
  __shared__ float As[3][16][APAD];

  const int tid   = threadIdx.x;
  const int lane  = tid & 31;
  const int wave  = tid >> 5;          // 0..7 -> N-tile
  const int mrow  = lane & 15;
  const int khalf = lane >> 4;
  const int ncol  = lane & 15;
  const int mbase = blockIdx.x * 16;
  const int nbase = wave * 16;

  // ---- stage A tiles: 3 mats x 16 rows x 128 floats = 1536 float4 slots ----
  const float* mats[3] = {aS, aD, xin};
#pragma unroll
  for (int p = 0; p < 6; ++p) {
    int idx = p * 256 + tid;           // 0..1535
    int mat = idx >> 9;                // /512 float4 per matrix
    int f4  = idx & 511;
    int r   = f4 >> 5;                 // row 0..15
    int c4  = (f4 & 31) * 4;           // col 0..124
    int row = mbase + r; if (row >= N) row = N - 1;
    async_copy16(mats[mat] + (size_t)row * D_FEAT + c4, &As[mat][r][c4]);
  }
  async_wait_all();
  __syncthreads();

  const float bias = ba[nbase + ncol] + bb[nbase + ncol] + bc[nbase + ncol];
  v8f acc;
#pragma unroll
  for (int r = 0; r < 8; ++r) acc[r] = bias;

#pragma unroll 4
  for (int ks = 0; ks < D_FEAT / 4; ++ks) {
    const int k0 = ks * 4 + khalf * 2;

    v2f a0 = *(const v2f*)(&As[0][mrow][k0]);     // ds_load_b64, bank-clean
    v2f b0;
    b0.x = Wa[(size_t)k0 * H_FEAT + nbase + ncol];
    b0.y = Wa[(size_t)(k0 + 1) * H_FEAT + nbase + ncol];
    acc = __builtin_amdgcn_wmma_f32_16x16x4_f32(false, a0, false, b0,
                                                (short)0, acc, false, false);

    v2f a1 = *(const v2f*)(&As[1][mrow][k0]);
    v2f b1;
    b1.x = Wb[(size_t)k0 * H_FEAT + nbase + ncol];
    b1.y = Wb[(size_t)(k0 + 1) * H_FEAT + nbase + ncol];
    acc = __builtin_amdgcn_wmma_f32_16x16x4_f32(false, a1, false, b1,
                                                (short)0, acc, false, false);

    v2f a2 = *(const v2f*)(&As[2][mrow][k0]);
    v2f b2;
    b2.x = Wc[(size_t)k0 * H_FEAT + nbase + ncol];
    b2.y = Wc[(size_t)(k0 + 1) * H_FEAT + nbase + ncol];
    acc = __builtin_amdgcn_wmma_f32_16x16x4_f32(false, a2, false, b2,
                                                (short)0, acc, false, false);
  }

#pragma unroll
  for (int r = 0; r < 8; ++r) {
    int orow = mbase + r + 8 * khalf;
    float v = acc[r];
    if (do_relu) v = fmaxf(v, 0.0f);
    if (orow < N) out[(size_t)orow * H_FEAT + nbase + ncol] = v;
  }
}

// ---------------------------------------------------------------------------
// Phase 3b (hidden layers): row L2 normalization, one wave per row.
// ---------------------------------------------------------------------------
__global__ void rownorm_kernel(float* __restrict__ x, int N) {
  const int wid  = blockIdx.x * (blockDim.x >> 5) + (threadIdx.x >> 5);
  const int lane = threadIdx.x & 31;
  if (wid >= N) return;
  float4 v = *(const float4*)(x + (size_t)wid * D_FEAT + lane * 4);
  float s = v.x * v.x + v.y * v.y + v.z * v.z + v.w * v.w;
#pragma unroll
  for (int off = 16; off > 0; off >>= 1) s += __shfl_xor(s, off, 32);
  const float inv = 1.0f / fmaxf(sqrtf(s), EPSV);
  v.x *= inv; v.y *= inv; v.z *= inv; v.w *= inv;
  *(float4*)(x + (size_t)wid * D_FEAT + lane * 4) = v;
}

// ---------------------------------------------------------------------------
// Phase 4: readout  out[N,64] = x @ W_ro + b_ro.
// Block = 128 thr (4 waves) = one 16-row M tile across all 4 N-tiles.
// ---------------------------------------------------------------------------
__global__ __launch_bounds__(128) void gemm_readout_kernel(
    const float* __restrict__ xin, const float* __restrict__ W,
    const float* __restrict__ b, float* __restrict__ out, int N) {
  __shared__ float Xs[16][APAD];

  const int tid   = threadIdx.x;
  const int lane  = tid & 31;
  const int wave  = tid >> 5;          // 0..3 -> N-tile
  const int mrow  = lane & 15;
  const int khalf = lane >> 4;
  const int ncol  = lane & 15;
  const int mbase = blockIdx.x * 16;
  const int nbase = wave * 16;

  // stage 16x128 x-tile: 512 float4 slots over 128 threads
#pragma unroll
  for (int p = 0; p < 4; ++p) {
    int idx = p * 128 + tid;           // 0..511
    int r   = idx >> 5;
    int c4  = (idx & 31) * 4;
    int row = mbase + r; if (row >= N) row = N - 1;
    async_copy16(xin + (size_t)row * D_FEAT + c4, &Xs[r][c4]);
  }
  async_wait_all();
  __syncthreads();

  const float bias = b[nbase + ncol];
  v8f acc;
#pragma unroll
  for (int r = 0; r < 8; ++r) acc[r] = bias;

#pragma unroll 4
  for (int ks = 0; ks < D_FEAT / 4; ++ks) {
    const int k0 = ks * 4 + khalf * 2;
    v2f a = *(const v2f*)(&Xs[mrow][k0]);
    v2f bf;
    bf.x = W[(size_t)k0 * O_FEAT + nbase + ncol];
    bf.y = W[(size_t)(k0 + 1) * O_FEAT + nbase + ncol];
    acc = __builtin_amdgcn_wmma_f32_16x16x4_f32(false, a, false, bf,
                                                (short)0, acc, false, false);
  }

#pragma unroll
  for (int r = 0; r < 8; ++r) {
    int orow = mbase + r + 8 * khalf;
    if (orow < N) out[(size_t)orow * O_FEAT + nbase + ncol] = acc[r];
  }
}

// ---------------------------------------------------------------------------
extern "C" void kernel_launch(void* const* d_in, const int* in_sizes, int n_in,
                              void* d_out, int out_size, void* d_ws, size_t ws_size,
                              hipStream_t stream) {
  const float*      x     = (const float*)d_in[0];
  const long long*  ei    = (const long long*)d_in[1];   // int64 per reference
  const float*      theta = (const float*)d_in[2];
  const float*      Ws2d  = (const float*)d_in[3];
  const float*      Wd2s  = (const float*)d_in[4];
  const float*      Wself = (const float*)d_in[5];
  const float*      bs2d  = (const float*)d_in[6];
  const float*      bd2s  = (const float*)d_in[7];
  const float*      bself = (const float*)d_in[8];
  const float*      W_ro  = (const float*)d_in[9];
  const float*      b_ro  = (const float*)d_in[10];

  const int N = in_sizes[0] / D_FEAT;
  const int E = in_sizes[2];
  const int L = in_sizes[3] / (D_FEAT * H_FEAT);
  const long long* src = ei;
  const long long* dst = ei + E;

  // workspace carve-up (floats)
  float* ws    = (float*)d_ws;
  float* degof = ws;
  float* degif = degof + N;
  float* degob = degif + N;
  float* degib = degob + N;
  float* wfn   = degib + N;
  float* wbn   = wfn + E;
  float* aS    = wbn + E;
  float* aD    = aS + (size_t)N * D_FEAT;
  float* xb1   = aD + (size_t)N * D_FEAT;
  float* xb2   = xb1 + (size_t)N * D_FEAT;

  const int TB = 256;

  // edge weights
  (void)hipMemsetAsync(degof, 0, sizeof(float) * 4 * (size_t)N, stream);
  deg_kernel<<<(E + TB - 1) / TB, TB, 0, stream>>>(src, dst, theta,
                                                   degof, degif, degob, degib, E);
  wnorm_kernel<<<(E + TB - 1) / TB, TB, 0, stream>>>(src, dst, theta,
                                                     degof, degif, degob, degib,
                                                     wfn, wbn, E);

  const int mtiles = (N + 15) / 16;
  const float* xcur = x;
  for (int l = 0; l < L; ++l) {
    (void)hipMemsetAsync(aS, 0, sizeof(float) * 2 * (size_t)N * D_FEAT, stream);
    long long tot = (long long)E * 32;
    agg_kernel<<<(int)((tot + TB - 1) / TB), TB, 0, stream>>>(xcur, src, dst,
                                                              wfn, wbn, aS, aD, E);
    float* hout = (l & 1) ? xb2 : xb1;
    int do_relu = (l != L - 1) ? 1 : 0;
    gemm_hidden_kernel<<<mtiles, 256, 0, stream>>>(
        aS, aD, xcur,
        Ws2d + (size_t)l * D_FEAT * H_FEAT,
        Wd2s + (size_t)l * D_FEAT * H_FEAT,
        Wself + (size_t)l * D_FEAT * H_FEAT,
        bs2d + (size_t)l * H_FEAT,
        bd2s + (size_t)l * H_FEAT,
        bself + (size_t)l * H_FEAT,
        hout, N, do_relu);
    if (do_relu) {
      int rowsPerBlk = TB / 32;
      rownorm_kernel<<<(N + rowsPerBlk - 1) / rowsPerBlk, TB, 0, stream>>>(hout, N);
    }
    xcur = hout;
  }

  gemm_readout_kernel<<<mtiles, 128, 0, stream>>>(
      xcur, W_ro, b_ro, (float*)d_out, N);
}